// TripletLoss_60541859005008
// MI455X (gfx1250) — compile-verified
//
#include <hip/hip_runtime.h>
#include <stdint.h>

typedef float v2f __attribute__((ext_vector_type(2)));
typedef float v4f __attribute__((ext_vector_type(4)));
typedef float v8f __attribute__((ext_vector_type(8)));

#define MARGIN 0.5f

// ---------------------------------------------------------------------------
// Kernel 1: squared L2 norm of each row.  One wave per row (8 waves / block).
// ---------------------------------------------------------------------------
__global__ void sq_norm_kernel(const float* __restrict__ x, float* __restrict__ sq,
                               int B, int D) {
    const int wave = threadIdx.x >> 5;
    const int lane = threadIdx.x & 31;
    const int row  = blockIdx.x * (blockDim.x >> 5) + wave;
    if (row >= B) return;
    const float* p = x + (long)row * D;
    float s = 0.f;
    for (int k = lane; k < D; k += 32) { float v = p[k]; s += v * v; }
    #pragma unroll
    for (int off = 16; off > 0; off >>= 1) s += __shfl_xor(s, off, 32);
    if (lane == 0) sq[row] = s;
}

// ---------------------------------------------------------------------------
// Async copy of one 8KB column tile (16 rows x 128 f32, contiguous in global)
// into LDS.  Each lane moves 4 x 16B chunks; the 24-bit IOFFSET is added to
// BOTH the global and the LDS address (CDNA5 ISA), so one
// per-lane cursor pair + constant offsets covers the whole tile.
// Tracked by ASYNCcnt.
// ---------------------------------------------------------------------------
__device__ __forceinline__ void issue_tile_copy(const char* gsrc, unsigned lds_dst) {
    asm volatile(
        "global_load_async_to_lds_b128 %0, %1, off\n\t"
        "global_load_async_to_lds_b128 %0, %1, off offset:512\n\t"
        "global_load_async_to_lds_b128 %0, %1, off offset:1024\n\t"
        "global_load_async_to_lds_b128 %0, %1, off offset:1536"
        :
        : "v"(lds_dst), "v"(gsrc)
        : "memory");
}

__device__ __forceinline__ void wait_async0() {
    asm volatile("s_wait_asynccnt 0x0" ::: "memory");
}

// ---------------------------------------------------------------------------
// Kernel 2: batch-hard mining.  4 waves / block, each wave owns a 16-anchor
// strip; all waves share a double-buffered 16x128 B tile in LDS fed by
// global_load_async_to_lds_b128.  Gram tile via V_WMMA_F32_16X16X4_F32
// (exact f32), K=128 in 32 steps.  K-order is permuted identically for A and
// B so b128 accesses work: per lane float4 j at float offset 8j+4h (h=lane/16);
// step 2j uses .xy, step 2j+1 uses .zw.  Mining in d^2-space (sqrt monotone).
// ---------------------------------------------------------------------------
__global__ void __launch_bounds__(128, 1)
triplet_tile_kernel(const float* __restrict__ x,
                    const int*   __restrict__ labels,
                    const float* __restrict__ sq,
                    float* __restrict__ row_loss,
                    float* __restrict__ row_valid,
                    int B) {
    constexpr int D  = 128;
    const float INF  = __builtin_inff();

    __shared__ float lds_b[2][16 * D];            // 2 x 8KB ping-pong

    const int tid    = threadIdx.x;
    const int lane   = tid & 31;
    const int w      = tid >> 5;                  // wave id 0..3
    const int lm     = lane & 15;
    const int h      = lane >> 4;
    const int m_base = blockIdx.x * 64 + w * 16;  // this wave's anchor strip
    const int m_row  = m_base + lm;

    // Per-lane async-copy cursors (lane's 16B chunk within the 8KB tile).
    const int chunk = w * 2048 + lane * 16;
    const char*    gcur = (const char*)x + chunk;                       // tile 0
    const unsigned lds0 = (unsigned)(uintptr_t)(&lds_b[0][0]) + chunk;  // buf 0

    // Preload the whole A tile: 16 b128 loads per lane (register resident).
    v4f areg[16];
    const float* pa = x + (long)m_row * D + 4 * h;
    #pragma unroll
    for (int j = 0; j < 16; ++j) areg[j] = *(const v4f*)(pa + 8 * j);

    // Anchor labels for the 8 C-rows this lane sees (M = r + 8h).
    int lab_m[8];
    #pragma unroll
    for (int r = 0; r < 8; ++r) lab_m[r] = labels[m_base + r + 8 * h];

    float hp[8], hn[8];                           // max(pos t) / min(neg t)
    #pragma unroll
    for (int r = 0; r < 8; ++r) { hp[r] = -INF; hn[r] = INF; }

    const int ntiles = B >> 4;

    // Prologue: tile 0 -> buffer 0.
    issue_tile_copy(gcur, lds0);
    wait_async0();
    __syncthreads();

    for (int jt = 0; jt < ntiles; ++jt) {
        // Kick off the next tile's copy into the other buffer (overlaps WMMA).
        if (jt + 1 < ntiles)
            issue_tile_copy(gcur + (size_t)(jt + 1) * 8192,
                            lds0 + ((unsigned)((jt + 1) & 1)) * 8192u);

        const int   n     = jt * 16 + lm;         // B column for this lane
        const int   lab_n = labels[n];
        const float sq_n  = sq[n];

        // Consume current buffer: 16 ds_load_b128 (in chunks of 4) + 32 WMMAs.
        const float* bs = &lds_b[jt & 1][lm * D + 4 * h];
        v8f c0 = {}; v8f c1 = {};
        #pragma unroll
        for (int jj = 0; jj < 4; ++jj) {
            v4f bt[4];
            #pragma unroll
            for (int q = 0; q < 4; ++q) bt[q] = *(const v4f*)(bs + 8 * (4 * jj + q));
            #pragma unroll
            for (int q = 0; q < 4; ++q) {
                const int j = 4 * jj + q;
                v2f a_lo = __builtin_shufflevector(areg[j], areg[j], 0, 1);
                v2f a_hi = __builtin_shufflevector(areg[j], areg[j], 2, 3);
                v2f b_lo = __builtin_shufflevector(bt[q], bt[q], 0, 1);
                v2f b_hi = __builtin_shufflevector(bt[q], bt[q], 2, 3);
                c0 = __builtin_amdgcn_wmma_f32_16x16x4_f32(false, a_lo, false, b_lo,
                                                           (short)0, c0, false, false);
                c1 = __builtin_amdgcn_wmma_f32_16x16x4_f32(false, a_hi, false, b_hi,
                                                           (short)0, c1, false, false);
            }
        }

        #pragma unroll
        for (int r = 0; r < 8; ++r) {
            const float dot = c0[r] + c1[r];
            const float t   = fmaf(-2.f, dot, sq_n);   // d^2 minus sq[m]
            const int   mr  = m_base + r + 8 * h;
            const bool same = (lab_n == lab_m[r]);
            if (same & (n != mr)) hp[r] = fmaxf(hp[r], t);
            if (!same)            hn[r] = fminf(hn[r], t);
        }

        // Next tile's copy must be complete (each wave waits for its own
        // chunks, the barrier makes all chunks visible to all waves) and all
        // waves must be done reading the current buffer before it is reused.
        wait_async0();
        __syncthreads();
    }

    // Reduce across the 16 lanes sharing each C-row (xor 1,2,4,8 stays in group).
    #pragma unroll
    for (int r = 0; r < 8; ++r) {
        #pragma unroll
        for (int off = 1; off < 16; off <<= 1) {
            hp[r] = fmaxf(hp[r], __shfl_xor(hp[r], off, 32));
            hn[r] = fminf(hn[r], __shfl_xor(hn[r], off, 32));
        }
    }

    // Lane 0 writes rows m_base+0..7, lane 16 writes m_base+8..15.
    if (lm == 0) {
        #pragma unroll
        for (int r = 0; r < 8; ++r) {
            const int   mr    = m_base + r + 8 * h;
            const float sqm   = sq[mr];
            const bool  valid = (hp[r] > -INF) && (hn[r] < INF);
            const float d2p   = fmaxf(sqm + hp[r], 0.f);
            const float d2n   = fmaxf(sqm + hn[r], 0.f);
            const float loss  = fmaxf(sqrtf(d2p) - sqrtf(d2n) + MARGIN, 0.f);
            row_loss[mr]  = valid ? loss : 0.f;
            row_valid[mr] = valid ? 1.f  : 0.f;
        }
    }
}

// ---------------------------------------------------------------------------
// Kernel 3: deterministic scalar reduction -> out[0].
// ---------------------------------------------------------------------------
__global__ void finalize_kernel(const float* __restrict__ row_loss,
                                const float* __restrict__ row_valid,
                                float* __restrict__ out, int B) {
    __shared__ float s_sum[256];
    __shared__ float s_cnt[256];
    float s = 0.f, c = 0.f;
    for (int i = threadIdx.x; i < B; i += blockDim.x) {
        s += row_loss[i];
        c += row_valid[i];
    }
    s_sum[threadIdx.x] = s;
    s_cnt[threadIdx.x] = c;
    __syncthreads();
    for (int off = blockDim.x >> 1; off > 0; off >>= 1) {
        if ((int)threadIdx.x < off) {
            s_sum[threadIdx.x] += s_sum[threadIdx.x + off];
            s_cnt[threadIdx.x] += s_cnt[threadIdx.x + off];
        }
        __syncthreads();
    }
    if (threadIdx.x == 0) out[0] = s_sum[0] / fmaxf(s_cnt[0], 1.f);
}

// ---------------------------------------------------------------------------
extern "C" void kernel_launch(void* const* d_in, const int* in_sizes, int n_in,
                              void* d_out, int out_size, void* d_ws, size_t ws_size,
                              hipStream_t stream) {
    const float* emb    = (const float*)d_in[0];   // [B, 128] f32
    const int*   labels = (const int*)d_in[1];     // [B] int
    const int B = in_sizes[1];
    const int D = in_sizes[0] / B;                 // 128

    float* ws        = (float*)d_ws;
    float* sq        = ws;                         // [B]
    float* row_loss  = ws + B;                     // [B]
    float* row_valid = ws + 2 * B;                 // [B]

    sq_norm_kernel<<<(B + 7) / 8, 256, 0, stream>>>(emb, sq, B, D);
    triplet_tile_kernel<<<B / 64, 128, 0, stream>>>(emb, labels, sq,
                                                    row_loss, row_valid, B);
    finalize_kernel<<<1, 256, 0, stream>>>(row_loss, row_valid, (float*)d_out, B);
}